// GCN_81286551044232
// MI455X (gfx1250) — compile-verified
//
#include <hip/hip_runtime.h>
#include <cstddef>

// GCN, 2 layers, fused norm/bias/relu/residual. Matrix ops via
// V_WMMA_F32_16X16X4_F32 (fp32 exact w.r.t. reference accumulation).

typedef __attribute__((ext_vector_type(2))) float v2f;
typedef __attribute__((ext_vector_type(8))) float v8f;

#define FEAT_D 128

// ---- degree accumulation (unweighted, per DGL norm='both') ----
__global__ void gcn_degree_kernel(const int* __restrict__ src, const int* __restrict__ dst,
                                  float* __restrict__ deg_s, float* __restrict__ deg_d, int E) {
  int i = blockIdx.x * blockDim.x + threadIdx.x;
  if (i < E) {
    atomicAdd(&deg_s[src[i]], 1.0f);
    atomicAdd(&deg_d[dst[i]], 1.0f);
  }
}

// ---- norm = rsqrt(max(deg, 1)) ----
__global__ void gcn_norm_kernel(const float* __restrict__ deg_s, const float* __restrict__ deg_d,
                                float* __restrict__ out_norm, float* __restrict__ in_norm, int N) {
  int i = blockIdx.x * blockDim.x + threadIdx.x;
  if (i < N) {
    out_norm[i] = rsqrtf(fmaxf(deg_s[i], 1.0f));
    in_norm[i]  = rsqrtf(fmaxf(deg_d[i], 1.0f));
  }
}

// ---- feat = h * out_norm[row], float4 vectorized ----
__global__ void gcn_scale_kernel(const float* __restrict__ h, const float* __restrict__ norm,
                                 float* __restrict__ feat, int n4) {
  int i = blockIdx.x * blockDim.x + threadIdx.x;
  if (i < n4) {
    float4 v = reinterpret_cast<const float4*>(h)[i];
    float s = norm[i >> 5];  // 32 float4 per 128-float row
    v.x *= s; v.y *= s; v.z *= s; v.w *= s;
    reinterpret_cast<float4*>(feat)[i] = v;
  }
}

// ---- one wave32 per edge: gather feat[src] (float4/lane), scatter-add into agg[dst] ----
__global__ void gcn_scatter_kernel(const float* __restrict__ feat, const float* __restrict__ ew,
                                   const int* __restrict__ src, const int* __restrict__ dst,
                                   float* __restrict__ agg, int E) {
  int gid = blockIdx.x * blockDim.x + threadIdx.x;
  int e = gid >> 5;
  int lane = gid & 31;
  if (e >= E) return;                      // wave-uniform
  int s = src[e];
  int d = dst[e];
  float w = ew[e];
  float4 v = reinterpret_cast<const float4*>(feat + (size_t)s * FEAT_D)[lane];
  float* out = agg + (size_t)d * FEAT_D + lane * 4;
  atomicAdd(out + 0, v.x * w);             // non-returning -> fire-and-forget L2 atomics
  atomicAdd(out + 1, v.y * w);
  atomicAdd(out + 2, v.z * w);
  atomicAdd(out + 3, v.w * w);
}

// ---- D = relu((agg*in_norm) @ W + b) + hprev, one wave per 16x16 tile ----
__global__ void gcn_gemm_kernel(const float* __restrict__ agg, const float* __restrict__ in_norm,
                                const float* __restrict__ W, const float* __restrict__ bias,
                                const float* __restrict__ hprev, float* __restrict__ hout, int N) {
  const int lane = threadIdx.x & 31;
  const int waveId = blockIdx.x * (blockDim.x >> 5) + (threadIdx.x >> 5);
  const int tilesN = FEAT_D / 16;          // 8 column tiles
  const int tile_m = waveId / tilesN;
  const int tile_n = waveId % tilesN;
  if (tile_m * 16 >= N) return;            // wave-uniform: EXEC stays all-1s for WMMA

  const int half = lane >> 4;              // 0 or 1
  const int l15  = lane & 15;
  int row_a = tile_m * 16 + l15;
  if (row_a >= N) row_a = N - 1;           // clamp loads; stores guarded below
  const int koff = half << 1;              // lanes 0-15: K={k,k+1}; lanes 16-31: K={k+2,k+3}
  const float anorm = in_norm[row_a];
  const int col_b = tile_n * 16 + l15;

  const float* aRow = agg + (size_t)row_a * FEAT_D;
  v8f acc = {};
#pragma unroll
  for (int k = 0; k < FEAT_D; k += 4) {
    v2f a, b;
    a.x = aRow[k + koff] * anorm;          // fold in_norm into A fragment
    a.y = aRow[k + koff + 1] * anorm;
    b.x = W[(size_t)(k + koff) * FEAT_D + col_b];
    b.y = W[(size_t)(k + koff + 1) * FEAT_D + col_b];
    acc = __builtin_amdgcn_wmma_f32_16x16x4_f32(
        /*neg_a=*/false, a, /*neg_b=*/false, b,
        /*c_mod=*/(short)0, acc, /*reuse_a=*/false, /*reuse_b=*/false);
  }

  const float bcol = bias[col_b];
  const int row_base = tile_m * 16 + (half << 3);   // VGPR r, lanes>=16 -> M += 8
#pragma unroll
  for (int r = 0; r < 8; ++r) {
    int row = row_base + r;
    if (row < N) {
      float v = acc[r] + bcol;
      v = v > 0.0f ? v : 0.0f;                       // relu
      hout[(size_t)row * FEAT_D + col_b] = v + hprev[(size_t)row * FEAT_D + col_b]; // residual
    }
  }
}

extern "C" void kernel_launch(void* const* d_in, const int* in_sizes, int n_in,
                              void* d_out, int out_size, void* d_ws, size_t ws_size,
                              hipStream_t stream) {
  const float* x  = (const float*)d_in[0];
  const float* ew = (const float*)d_in[1];
  const int*   src = (const int*)d_in[2];
  const int*   dst = (const int*)d_in[3];
  const float* W1 = (const float*)d_in[4];
  const float* b1 = (const float*)d_in[5];
  const float* W2 = (const float*)d_in[6];
  const float* b2 = (const float*)d_in[7];

  const int N = in_sizes[0] / FEAT_D;
  const int E = in_sizes[1];

  // workspace layout (floats): [out_norm N][in_norm N][feat N*D][agg N*D]
  // degree scratch aliases the feat region (consumed before feat is written).
  float* ws = (float*)d_ws;
  float* out_norm = ws;
  float* in_norm  = ws + N;
  float* feat = ws + 2 * (size_t)N;
  float* agg  = feat + (size_t)N * FEAT_D;
  float* deg_s = feat;
  float* deg_d = feat + N;

  float* h1 = (float*)d_out;  // layer-1 output lives in d_out; layer 2 updates in place

  const int T = 256;          // 8 wave32 per block
  const int n4 = N * (FEAT_D / 4);
  const int tiles = ((N + 15) / 16) * (FEAT_D / 16);
  const int gemmBlocks = (tiles + (T / 32) - 1) / (T / 32);
  const long long scatterThreads = (long long)E * 32;
  const int scatterBlocks = (int)((scatterThreads + T - 1) / T);

  // degrees + norms (once; shared by both layers)
  hipMemsetAsync(deg_s, 0, 2 * (size_t)N * sizeof(float), stream);
  gcn_degree_kernel<<<(E + T - 1) / T, T, 0, stream>>>(src, dst, deg_s, deg_d, E);
  gcn_norm_kernel<<<(N + T - 1) / T, T, 0, stream>>>(deg_s, deg_d, out_norm, in_norm, N);

  // ---- layer 1 (h = x) ----
  gcn_scale_kernel<<<(n4 + T - 1) / T, T, 0, stream>>>(x, out_norm, feat, n4);
  hipMemsetAsync(agg, 0, (size_t)N * FEAT_D * sizeof(float), stream);
  gcn_scatter_kernel<<<scatterBlocks, T, 0, stream>>>(feat, ew, src, dst, agg, E);
  gcn_gemm_kernel<<<gemmBlocks, T, 0, stream>>>(agg, in_norm, W1, b1, x, h1, N);

  // ---- layer 2 (h = h1, output in place) ----
  gcn_scale_kernel<<<(n4 + T - 1) / T, T, 0, stream>>>(h1, out_norm, feat, n4);
  hipMemsetAsync(agg, 0, (size_t)N * FEAT_D * sizeof(float), stream);
  gcn_scatter_kernel<<<scatterBlocks, T, 0, stream>>>(feat, ew, src, dst, agg, E);
  gcn_gemm_kernel<<<gemmBlocks, T, 0, stream>>>(agg, in_norm, W2, b2, h1, h1, N);
}